// Dynamic_System_61349312856238
// MI455X (gfx1250) — compile-verified
//
#include <hip/hip_runtime.h>

// MI455X / gfx1250, wave32. Fully fused batched-MLP dynamics solve.
// Pre-pass packs all per-K-chunk weights (layer-1 scalar section + layer-2
// WMMA B-tiles in ISA lane layout, zero-padded) into one contiguous record
// per chunk in d_ws, so the main loop runs off a single base pointer:
// scalar s_load_b128 for uniform weights, one coalesced b64 per B operand,
// and v_wmma_f32_16x16x4_f32 for the K=512 layer-2 GEMMs.

typedef float v2f __attribute__((ext_vector_type(2)));
typedef float v8f __attribute__((ext_vector_type(8)));

#define HID 512
#define NCHUNK (HID / 4)
#define REC 264  // floats per chunk record: 72 scalar + 3*64 B-operand

// ---------------- pre-pass: pack weights into chunk records ----------------
__global__ __launch_bounds__(64) void pack_weights(
    const float* __restrict__ W1m, const float* __restrict__ b1m,
    const float* __restrict__ W2m,
    const float* __restrict__ W1g, const float* __restrict__ b1g,
    const float* __restrict__ W2g,
    const float* __restrict__ W1k, const float* __restrict__ b1k,
    const float* __restrict__ W2k,
    float* __restrict__ pk)
{
  const int c  = blockIdx.x;   // K-chunk 0..127
  const int t  = threadIdx.x;  // 0..63
  const int k4 = c * 4;
  float* rec = pk + c * REC;

  // scalar section [0..71]: W1m(12) b1m(4) W1g(12) b1g(4) W1k(36) b1k(4)
  for (int idx = t; idx < 72; idx += 64) {
    float v;
    if (idx < 12)      v = W1m[(idx >> 2) * HID + k4 + (idx & 3)];
    else if (idx < 16) v = b1m[k4 + (idx & 3)];
    else if (idx < 28) { const int j = idx - 16; v = W1g[(j >> 2) * HID + k4 + (j & 3)]; }
    else if (idx < 32) v = b1g[k4 + (idx & 3)];
    else if (idx < 68) { const int j = idx - 32; v = W1k[(j >> 2) * HID + k4 + (j & 3)]; }
    else               v = b1k[k4 + (idx & 3)];
    rec[idx] = v;
  }

  // vector section: B operands in WMMA f32 16x16x4 B layout, zero padded.
  // element t: lane L = t>>1, component = t&1 -> K = k4 + (L>=16?2:0) + comp, N = L&15
  {
    const int L    = t >> 1;
    const int comp = t & 1;
    const int k    = k4 + ((L >= 16) ? 2 : 0) + comp;
    const int n    = L & 15;
    rec[72  + t] = (n < 9) ? W2m[k * 9 + n] : 0.0f;
    rec[136 + t] = (n < 3) ? W2g[k * 3 + n] : 0.0f;
    rec[200 + t] = (n < 3) ? W2k[k * 3 + n] : 0.0f;
  }
}

// ---------------- main fused kernel ----------------
__device__ __forceinline__ float fast_tanhf(float x) {
  // tanh(x) = (e^{2x}-1) * rcp(e^{2x}+1); clamp avoids inf*0 at overflow.
  x = fminf(x, 20.0f);
  const float e = __builtin_amdgcn_exp2f(x * 2.8853900817779268f); // 2/ln(2)
  return (e - 1.0f) * __builtin_amdgcn_rcpf(e + 1.0f);
}

__device__ __forceinline__ void pick2(const float4 v, bool hi, float& a, float& b) {
  a = hi ? v.z : v.x;
  b = hi ? v.w : v.y;
}

__global__ __launch_bounds__(256) void dyn_fused(
    const float* __restrict__ q,   const float* __restrict__ q_dot,
    const float* __restrict__ s,   const float* __restrict__ s_Ddot,
    const float* __restrict__ tau, const float* __restrict__ fv,
    const float* __restrict__ fc,
    const float* __restrict__ b2m, const float* __restrict__ b2g,
    const float* __restrict__ b2k,
    const float* __restrict__ pk,
    float* __restrict__ out, int nrows)
{
  __shared__ float tile[8][16][17]; // [wave][row][15 used cols], stride 17: conflict-free

  const int  lane = threadIdx.x & 31;
  const int  wid  = threadIdx.x >> 5;
  const int  n    = lane & 15;
  const bool hi   = lane >= 16;
  const int  row0 = (blockIdx.x * 8 + wid) * 16;
  const int  r    = row0 + n;
  const bool valid = r < nrows;

  float xq[9];
#pragma unroll
  for (int j = 0; j < 3; ++j) {
    xq[j]     = valid ? q[r * 3 + j]      : 0.0f;
    xq[3 + j] = valid ? s[r * 3 + j]      : 0.0f;
    xq[6 + j] = valid ? s_Ddot[r * 3 + j] : 0.0f;
  }

  // C accumulators start as broadcast layer-2 bias.
  const float bmb = (n < 9) ? b2m[n] : 0.0f;
  const float bgb = (n < 3) ? b2g[n] : 0.0f;
  const float bkb = (n < 3) ? b2k[n] : 0.0f;
  v8f cm, cg, ck;
#pragma unroll
  for (int v = 0; v < 8; ++v) { cm[v] = bmb; cg[v] = bgb; ck[v] = bkb; }

#pragma unroll 1
  for (int c = 0; c < NCHUNK; ++c) {
    const float* rec = pk + c * REC;
    const float4* r4 = reinterpret_cast<const float4*>(rec); // uniform -> s_load_b128

    // B operands: one VGPR address, 3 coalesced b64 loads (pre-padded).
    const float* vb = rec + 72 + (lane << 1);
    const v2f bm = *reinterpret_cast<const v2f*>(vb);
    const v2f bg = *reinterpret_cast<const v2f*>(vb + 64);
    const v2f bk = *reinterpret_cast<const v2f*>(vb + 128);

    float wa, wb;

    // ---- mass net layer 1 ----
    float za, zb;
    pick2(r4[3], hi, za, zb);
#pragma unroll
    for (int i = 0; i < 3; ++i) {
      pick2(r4[i], hi, wa, wb);
      za = fmaf(xq[i], wa, za); zb = fmaf(xq[i], wb, zb);
    }
    v2f am; am.x = fast_tanhf(za); am.y = fast_tanhf(zb);
    cm = __builtin_amdgcn_wmma_f32_16x16x4_f32(false, am, false, bm, (short)0, cm, false, false);

    // ---- gravity net layer 1 ----
    pick2(r4[7], hi, za, zb);
#pragma unroll
    for (int i = 0; i < 3; ++i) {
      pick2(r4[4 + i], hi, wa, wb);
      za = fmaf(xq[i], wa, za); zb = fmaf(xq[i], wb, zb);
    }
    v2f ag; ag.x = fast_tanhf(za); ag.y = fast_tanhf(zb);
    cg = __builtin_amdgcn_wmma_f32_16x16x4_f32(false, ag, false, bg, (short)0, cg, false, false);

    // ---- KAB net layer 1 ----
    pick2(r4[17], hi, za, zb);
#pragma unroll
    for (int i = 0; i < 9; ++i) {
      pick2(r4[8 + i], hi, wa, wb);
      za = fmaf(xq[i], wa, za); zb = fmaf(xq[i], wb, zb);
    }
    v2f ak; ak.x = fast_tanhf(za); ak.y = fast_tanhf(zb);
    ck = __builtin_amdgcn_wmma_f32_16x16x4_f32(false, ak, false, bk, (short)0, ck, false, false);
  }

  // Scatter D tiles to LDS: lane holds (M = v (+8 if hi), N = n).
#pragma unroll
  for (int v = 0; v < 8; ++v) {
    const int m = hi ? v + 8 : v;
    if (n < 9) tile[wid][m][n] = cm[v];
    if (n < 3) { tile[wid][m][9 + n] = cg[v]; tile[wid][m][12 + n] = ck[v]; }
  }
  __syncthreads();

  // ---- per-sample epilogue: rhs assembly + symmetric 3x3 solve ----
  if (lane < 16 && (row0 + lane) < nrows) {
    const int rr = row0 + lane;
    float Mr[9];
#pragma unroll
    for (int i = 0; i < 9; ++i) Mr[i] = tile[wid][lane][i];

    float rhs[3];
#pragma unroll
    for (int j = 0; j < 3; ++j) {
      const float qd = q_dot[rr * 3 + j];
      const float sg = (qd > 0.0f) ? 1.0f : ((qd < 0.0f) ? -1.0f : 0.0f);
      rhs[j] = tau[rr * 3 + j] - tile[wid][lane][9 + j] + tile[wid][lane][12 + j]
               - fv[j] * qd - fc[j] * sg;
    }

    // M = Mraw * Mraw^T + 0.1 I  (symmetric)
    const float m00 = fmaf(Mr[0], Mr[0], fmaf(Mr[1], Mr[1], fmaf(Mr[2], Mr[2], 0.1f)));
    const float m01 = fmaf(Mr[0], Mr[3], fmaf(Mr[1], Mr[4], Mr[2] * Mr[5]));
    const float m02 = fmaf(Mr[0], Mr[6], fmaf(Mr[1], Mr[7], Mr[2] * Mr[8]));
    const float m11 = fmaf(Mr[3], Mr[3], fmaf(Mr[4], Mr[4], fmaf(Mr[5], Mr[5], 0.1f)));
    const float m12 = fmaf(Mr[3], Mr[6], fmaf(Mr[4], Mr[7], Mr[5] * Mr[8]));
    const float m22 = fmaf(Mr[6], Mr[6], fmaf(Mr[7], Mr[7], fmaf(Mr[8], Mr[8], 0.1f)));

    // adjugate (symmetric) + Cramer
    const float a00 = m11 * m22 - m12 * m12;
    const float a01 = m02 * m12 - m01 * m22;
    const float a02 = m01 * m12 - m02 * m11;
    const float a11 = m00 * m22 - m02 * m02;
    const float a12 = m01 * m02 - m00 * m12;
    const float a22 = m00 * m11 - m01 * m01;
    const float det = fmaf(m00, a00, fmaf(m01, a01, m02 * a02));
    const float id  = 1.0f / det;

    out[rr * 3 + 0] = (a00 * rhs[0] + a01 * rhs[1] + a02 * rhs[2]) * id;
    out[rr * 3 + 1] = (a01 * rhs[0] + a11 * rhs[1] + a12 * rhs[2]) * id;
    out[rr * 3 + 2] = (a02 * rhs[0] + a12 * rhs[1] + a22 * rhs[2]) * id;
  }
}

extern "C" void kernel_launch(void* const* d_in, const int* in_sizes, int n_in,
                              void* d_out, int out_size, void* d_ws, size_t ws_size,
                              hipStream_t stream) {
  const float* q      = (const float*)d_in[0];
  const float* q_dot  = (const float*)d_in[1];
  const float* s      = (const float*)d_in[2];
  const float* s_Ddot = (const float*)d_in[3];
  const float* tau    = (const float*)d_in[4];
  const float* fv     = (const float*)d_in[5];
  const float* fc     = (const float*)d_in[6];
  const float* W1m    = (const float*)d_in[7];
  const float* b1m    = (const float*)d_in[8];
  const float* W2m    = (const float*)d_in[9];
  const float* b2m    = (const float*)d_in[10];
  const float* W1g    = (const float*)d_in[11];
  const float* b1g    = (const float*)d_in[12];
  const float* W2g    = (const float*)d_in[13];
  const float* b2g    = (const float*)d_in[14];
  const float* W1k    = (const float*)d_in[15];
  const float* b1k    = (const float*)d_in[16];
  const float* W2k    = (const float*)d_in[17];
  const float* b2k    = (const float*)d_in[18];

  float* pk = (float*)d_ws;  // 128 * 264 * 4 = 135168 bytes

  pack_weights<<<NCHUNK, 64, 0, stream>>>(W1m, b1m, W2m, W1g, b1g, W2g,
                                          W1k, b1k, W2k, pk);

  const int nrows  = in_sizes[0] / 3;    // B
  const int tiles  = (nrows + 15) / 16;  // 16 samples per wave
  const int blocks = (tiles + 7) / 8;    // 8 waves per 256-thread block

  dyn_fused<<<blocks, 256, 0, stream>>>(q, q_dot, s, s_Ddot, tau, fv, fc,
                                        b2m, b2g, b2k, pk,
                                        (float*)d_out, nrows);
}